// DechirpSTFT_35656818491856
// MI455X (gfx1250) — compile-verified
//
#include <hip/hip_runtime.h>
#include <math.h>

// ---------------- problem constants ----------------
#define KLEN   1024          // window / FFT length (K_TAU == K since R_WARP==1)
#define HOP    512
#define NBATCH 4
#define NSAMP  262144
#define NWIN   511           // (262144-1024)/512 + 1
#define NCHIRP 16
#define NFREQ  513           // rfft bins
#define NB     1056          // GEMM N: 66 tiles of 16 (cos cols 0..527, -sin cols 528..1055)
#define NTILES 66
#define EPSF   1e-8f
#define TWO_PI 6.28318530717958647692f

typedef __attribute__((ext_vector_type(2))) float v2f;
typedef __attribute__((ext_vector_type(8))) float v8f;

// ---------------- init kernel 1: per-chirp resampling tables ----------------
__global__ void dechirp_init_tables(const float* __restrict__ dlnf,
                                    int*   __restrict__ idxLo,
                                    float* __restrict__ frac,
                                    float* __restrict__ jac) {
    int i = blockIdx.x * blockDim.x + threadIdx.x;   // 0 .. 16*1024-1
    if (i >= NCHIRP * KLEN) return;
    int c = i >> 10;
    int t = i & (KLEN - 1);

    float beta  = 2.0f * dlnf[c];
    bool  small = fabsf(beta) < EPSF;
    float bs    = small ? EPSF : beta;
    float tau   = 2.0f * (float)t * (1.0f / (float)KLEN) - 1.0f;
    float e2bm1 = expf(2.0f * bs) - 1.0f;

    float ts = log1pf((tau + 1.0f) * 0.5f * e2bm1) / bs - 1.0f;
    if (small) ts = tau;
    // tau_mid == 0.0 exactly (2*(K/2)/K - 1)
    float tm = log1pf(0.5f * e2bm1) / bs - 1.0f;
    if (small) tm = 0.0f;
    float j = expf(-bs * (ts - tm));
    if (small) j = 1.0f;

    float idx = 512.0f * (ts + 1.0f);
    int   il  = (int)idx;                 // truncate toward zero, like astype(int32)
    il = il < 0 ? 0 : (il > KLEN - 2 ? KLEN - 2 : il);

    idxLo[i] = il;
    frac[i]  = idx - (float)il;
    jac[i]   = j;
}

// ---------------- init kernel 2: K-major DFT table ----------------
// BT[n][k], n in [0,1056): n<528 -> cos(2*pi*n*k/1024); else -> -sin(2*pi*(n-528)*k/1024)
__global__ void dechirp_init_dft(float* __restrict__ BT) {
    int i = blockIdx.x * blockDim.x + threadIdx.x;   // 0 .. 1056*1024-1
    if (i >= NB * KLEN) return;
    int n = i >> 10;
    int k = i & (KLEN - 1);
    int f = (n < 528) ? n : (n - 528);
    int m = (f * k) & (KLEN - 1);                    // exact angle reduction
    float ang = (float)m * (TWO_PI / (float)KLEN);
    BT[i] = (n < 528) ? cosf(ang) : -sinf(ang);
}

// ---------------- main kernel: resample into LDS A-tile, fp32-WMMA DFT GEMM ----------------
__global__ __launch_bounds__(256) void dechirp_stft_wmma(
    const float* __restrict__ x,
    const int*   __restrict__ idxLo,
    const float* __restrict__ frac,
    const float* __restrict__ jac,
    const float* __restrict__ BT,
    float*       __restrict__ out)   // interleaved complex64: [b][w][c][f]{re,im}
{
    __shared__ float sX[KLEN];               // windowed samples, 4 KB
    __shared__ float sA[NCHIRP * KLEN];      // A-tile (16 rows x 1024), 64 KB

    const int bw  = blockIdx.x;              // 0 .. 2043  == b*511 + w
    const int b   = bw / NWIN;
    const int w   = bw - b * NWIN;
    const int tid = threadIdx.x;

    // 1) windowed samples for this (b, w)
    const float* xrow = x + (size_t)b * NSAMP + (size_t)w * HOP;
    for (int t = tid; t < KLEN; t += 256) {
        float win = 0.5f * (1.0f - cosf(TWO_PI * (float)t * (1.0f / (float)KLEN)));
        sX[t] = xrow[t] * win;
    }
    __syncthreads();

    // 2) resampled A-tile: row = chirp, col = tau index
    for (int i = tid; i < NCHIRP * KLEN; i += 256) {
        int   il = idxLo[i];
        float fr = frac[i];
        float lo = sX[il];
        float hi = sX[il + 1];
        sA[i] = (lo * (1.0f - fr) + hi * fr) * jac[i];
    }
    __syncthreads();

    // 3) DFT as GEMM with V_WMMA_F32_16X16X4_F32, accumulator-chained over K.
    //    A fragment: lane<16 -> (M=lane,   K={k0,k0+1});  lane>=16 -> (M=lane-16, K={k0+2,k0+3})
    //    B fragment (K-major table): same per-lane contiguous K-pair at column n = nt*16 + (lane&15)
    const int lane = tid & 31;
    const int wave = tid >> 5;
    const int mrow = lane & 15;
    const int koff = (lane >> 4) << 1;                 // 0 or 2
    const int mhi  = (lane >> 4) << 3;                 // D: lanes 16-31 hold M = r+8
    const float* aBase = &sA[mrow * KLEN + koff];

    for (int nt = wave; nt < NTILES; nt += 8) {
        const int    ncol  = nt * 16 + mrow;           // this lane's N column
        const float* bBase = BT + (size_t)ncol * KLEN + koff;

        v8f acc = {};
        #pragma unroll 4
        for (int k0 = 0; k0 < KLEN; k0 += 4) {
            v2f a  = *(const v2f*)(aBase + k0);        // ds_load_b64 from LDS
            v2f bb = *(const v2f*)(bBase + k0);        // global_load_b64 (L2-resident table)
            acc = __builtin_amdgcn_wmma_f32_16x16x4_f32(
                /*neg_a=*/false, a, /*neg_b=*/false, bb,
                /*c_mod=*/(short)0, acc, /*reuse_a=*/false, /*reuse_b=*/false);
        }

        // 4) scatter D. Predicate depends only on ncol (lane), not r: hoist it so the
        //    8 per-chirp stores are unconditional inside one divergent region.
        //    Non-temporal: the 134 MB result stream must not evict the DFT table from L2.
        const size_t tileRow = (size_t)bw * NCHIRP + mhi;   // first chirp row for this lane half
        if (ncol < NFREQ) {
            float* p = out + (tileRow * NFREQ + ncol) * 2;  // real parts, stride 2*NFREQ floats
            #pragma unroll
            for (int r = 0; r < 8; ++r)
                __builtin_nontemporal_store(acc[r], p + (size_t)r * (2 * NFREQ));
        } else if (ncol >= 528 && ncol < 528 + NFREQ) {
            float* p = out + (tileRow * NFREQ + (ncol - 528)) * 2 + 1;  // imag parts
            #pragma unroll
            for (int r = 0; r < 8; ++r)
                __builtin_nontemporal_store(acc[r], p + (size_t)r * (2 * NFREQ));
        }
    }
}

// ---------------- launch ----------------
extern "C" void kernel_launch(void* const* d_in, const int* in_sizes, int n_in,
                              void* d_out, int out_size, void* d_ws, size_t ws_size,
                              hipStream_t stream) {
    const float* x    = (const float*)d_in[0];
    const float* dlnf = (const float*)d_in[1];
    // d_in[2] = n_hann_splits (always 1) — unused

    char*  ws    = (char*)d_ws;
    int*   idxLo = (int*)  (ws);                       //  64 KB
    float* frac  = (float*)(ws + (64u  << 10));        //  64 KB
    float* jac   = (float*)(ws + (128u << 10));        //  64 KB
    float* BT    = (float*)(ws + (192u << 10));        // 1056*1024*4 = 4.33 MB

    dechirp_init_tables<<<(NCHIRP * KLEN) / 256, 256, 0, stream>>>(dlnf, idxLo, frac, jac);
    dechirp_init_dft<<<(NB * KLEN + 255) / 256, 256, 0, stream>>>(BT);
    dechirp_stft_wmma<<<NBATCH * NWIN, 256, 0, stream>>>(x, idxLo, frac, jac, BT, (float*)d_out);
}